// SelfAttentionBlock_16234976379508
// MI455X (gfx1250) — compile-verified
//
#include <hip/hip_runtime.h>
#include <cstddef>

typedef __attribute__((ext_vector_type(2))) float v2f;
typedef __attribute__((ext_vector_type(8))) float v8f;

// Problem constants
constexpr int Bn   = 8;    // batch
constexpr int Hdim = 128;
constexpr int Wdim = 128;
constexpr int NCH  = 16;   // channels per group-pair (2 groups x 8)
// Tile
constexpr int TH   = 16;
constexpr int TW   = 8;
constexpr int HTH  = TH + 2;        // halo tile height
constexpr int HTW  = TW + 2;        // halo tile width
constexpr int HALO = HTH * HTW;     // 180 halo positions
constexpr int NB   = (HALO + 15) / 16; // 12 WMMA position-batches

// block-diagonal 16x16 weight built from two 8x8 group weight matrices
__device__ __forceinline__ float wblk(const float* __restrict__ Wt, int gp, int m, int kk) {
  if (m < 8)  return (kk < 8)  ? Wt[(2 * gp) * 64 + m * 8 + kk] : 0.0f;
  else        return (kk >= 8) ? Wt[(2 * gp + 1) * 64 + (m - 8) * 8 + (kk - 8)] : 0.0f;
}

__global__ __launch_bounds__(256) void sasa_fused_kernel(
    const float* __restrict__ x,  const float* __restrict__ Wq,
    const float* __restrict__ Wk, const float* __restrict__ Wv,
    const float* __restrict__ hE, const float* __restrict__ wE,
    float* __restrict__ out) {
  __shared__ float xt[HALO * NCH];
  __shared__ float qt[HALO * NCH];
  __shared__ float kt[HALO * NCH];
  __shared__ float vt[HALO * NCH];
  __shared__ float emb[NCH * 9];

  const int tid = threadIdx.x;
  const int bx  = blockIdx.x;          // W / TW
  const int by  = blockIdx.y;          // H / TH
  const int bz  = blockIdx.z;          // b * 4 + group_pair
  const int gp  = bz & 3;
  const int b   = bz >> 2;
  const int x0  = bx * TW, y0 = by * TH;
  const int cbase = gp * NCH;          // first global channel of this pair

  // ---- Phase 1: stage x halo tile into LDS (coalesced over x), build emb ----
  for (int idx = tid; idx < HALO * NCH; idx += 256) {
    int ch = idx / HALO;
    int p  = idx - ch * HALO;
    int hy = p / HTW, hx = p - hy * HTW;
    int gy = y0 + hy - 1, gx = x0 + hx - 1;
    float v = 0.0f;
    if (gy >= 0 && gy < Hdim && gx >= 0 && gx < Wdim)
      v = x[(((size_t)b * 64 + cbase + ch) * Hdim + gy) * Wdim + gx];
    xt[p * NCH + ch] = v;
  }
  if (tid < NCH * 9) {
    int ch = tid / 9, w9 = tid - ch * 9;
    int i = w9 / 3, j = w9 - i * 3;
    int g  = 2 * gp + (ch >> 3);
    int cg = ch & 7;
    emb[ch * 9 + w9] = (cg < 4) ? hE[(g * 4 + cg) * 3 + i]
                                : wE[(g * 4 + (cg - 4)) * 3 + j];
  }

  // ---- per-lane WMMA A-fragments (ISA 16x4 f32 A layout) ----
  const int ln = tid & 31;
  const int m  = ln & 15;     // A row / B,D column
  const int kh = ln >> 4;     // lane-half: K offset 0/1 vs 2/3 within each chunk
  v2f aq[4], ak[4], av[4];
  for (int c = 0; c < 4; ++c) {
    int k0 = 4 * c + 2 * kh;
    aq[c][0] = wblk(Wq, gp, m, k0); aq[c][1] = wblk(Wq, gp, m, k0 + 1);
    ak[c][0] = wblk(Wk, gp, m, k0); ak[c][1] = wblk(Wk, gp, m, k0 + 1);
    av[c][0] = wblk(Wv, gp, m, k0); av[c][1] = wblk(Wv, gp, m, k0 + 1);
  }
  __syncthreads();

  // ---- Phase 2: grouped 1x1 convs via V_WMMA_F32_16X16X4_F32 ----
  // D(16 out-ch x 16 pos) = sum_c A_c(16x4 blockdiag W) * B_c(4x16 x-slice)
  const int wv = tid >> 5;
  for (int batch = wv; batch < NB; batch += 8) {
    int pos = batch * 16 + m;                    // column n = lane&15
    int pc  = (pos < HALO) ? pos : (HALO - 1);   // clamp loads, keep EXEC full
    v2f bb[4];
    for (int c = 0; c < 4; ++c) {
      int k0 = 4 * c + 2 * kh;
      bb[c] = *(const v2f*)&xt[pc * NCH + k0];   // 8B-aligned (k0 even)
    }
    v8f cq = {}, ck = {}, cv = {};
    for (int c = 0; c < 4; ++c) {
      cq = __builtin_amdgcn_wmma_f32_16x16x4_f32(false, aq[c], false, bb[c],
                                                 (short)0, cq, false, false);
      ck = __builtin_amdgcn_wmma_f32_16x16x4_f32(false, ak[c], false, bb[c],
                                                 (short)0, ck, false, false);
      cv = __builtin_amdgcn_wmma_f32_16x16x4_f32(false, av[c], false, bb[c],
                                                 (short)0, cv, false, false);
    }
    if (pos < HALO) {  // D layout: VGPR r -> out-ch r + 8*kh at column pos
      float* q8 = &qt[pos * NCH + 8 * kh];
      float* k8 = &kt[pos * NCH + 8 * kh];
      float* v8 = &vt[pos * NCH + 8 * kh];
      #pragma unroll
      for (int r = 0; r < 8; ++r) { q8[r] = cq[r]; k8[r] = ck[r]; v8[r] = cv[r]; }
    }
  }
  __syncthreads();

  // ---- Phase 3: per-channel 3x3 softmax attention, coalesced store ----
  #pragma unroll
  for (int it = 0; it < (TH * TW * NCH) / 256; ++it) {
    int oi = it * 256 + tid;
    int ch = oi / (TH * TW);
    int p  = oi - ch * (TH * TW);
    int py = p / TW, px = p - py * TW;
    float q = qt[((py + 1) * HTW + (px + 1)) * NCH + ch];
    float lg[9];
    float mx = -1e30f;
    #pragma unroll
    for (int w9 = 0; w9 < 9; ++w9) {
      int i = w9 / 3, j = w9 - 3 * i;
      float kv = kt[((py + i) * HTW + (px + j)) * NCH + ch] + emb[ch * 9 + w9];
      float l = q * kv;
      lg[w9] = l;
      mx = fmaxf(mx, l);
    }
    float s = 0.0f, acc = 0.0f;
    #pragma unroll
    for (int w9 = 0; w9 < 9; ++w9) {
      int i = w9 / 3, j = w9 - 3 * i;
      float e = __expf(lg[w9] - mx);
      s   += e;
      acc += e * vt[((py + i) * HTW + (px + j)) * NCH + ch];
    }
    out[(((size_t)b * 64 + cbase + ch) * Hdim + (y0 + py)) * Wdim + (x0 + px)] = acc / s;
  }
}

extern "C" void kernel_launch(void* const* d_in, const int* in_sizes, int n_in,
                              void* d_out, int out_size, void* d_ws, size_t ws_size,
                              hipStream_t stream) {
  (void)in_sizes; (void)n_in; (void)d_ws; (void)ws_size; (void)out_size;
  const float* x  = (const float*)d_in[0];
  const float* Wq = (const float*)d_in[1];
  const float* Wk = (const float*)d_in[2];
  const float* Wv = (const float*)d_in[3];
  const float* hE = (const float*)d_in[4];
  const float* wE = (const float*)d_in[5];
  float* out = (float*)d_out;

  dim3 grid(Wdim / TW, Hdim / TH, Bn * 4);  // 16 x 8 x 32 = 4096 blocks
  sasa_fused_kernel<<<grid, 256, 0, stream>>>(x, Wq, Wk, Wv, hE, wE, out);
}